// MHA_36249523978370
// MI455X (gfx1250) — compile-verified
//
#include <hip/hip_runtime.h>
#include <math.h>

#define B_ 32
#define C_ 1024
#define L_ 2048
#define ROWS (B_ * C_)            // 32768 rows of length L_
#define WAVES_PER_BLOCK 8
#define THREADS (WAVES_PER_BLOCK * 32)

typedef unsigned int u32x4 __attribute__((ext_vector_type(4)));
typedef int i32x8 __attribute__((ext_vector_type(8)));
typedef int i32x4 __attribute__((ext_vector_type(4)));

#if __has_builtin(__builtin_amdgcn_tensor_load_to_lds)
#define HAVE_TDM 1
#else
#define HAVE_TDM 0
#endif

// ---------------------------------------------------------------------------
// Kernel 1: per-(b,c) row reduction over L: max and mean of 2048 f32.
// One wave per row. TDM DMA's the 8KB row into LDS (TENSORcnt), then the wave
// reduces out of LDS. Fallback: direct global_load_b128 streaming.
// ---------------------------------------------------------------------------
__global__ __launch_bounds__(THREADS) void pool_kernel(
    const float* __restrict__ x, float* __restrict__ mx, float* __restrict__ av) {
  __shared__ float lds[WAVES_PER_BLOCK * L_];   // 64 KB
  const int lane = (int)(threadIdx.x & 31u);
  const int wave = __builtin_amdgcn_readfirstlane((int)(threadIdx.x >> 5));
  const int row  = (int)blockIdx.x * WAVES_PER_BLOCK + wave;

#if HAVE_TDM
  {
    unsigned long long ga =
        (unsigned long long)(const void*)(x + (size_t)row * L_);
    unsigned int lds_base =
        (unsigned int)(unsigned long long)(const void*)&lds[wave * L_];

    // D# group 0 (128b): count=1 | lds_addr | global_addr[56:0] | type=2
    u32x4 g0;
    g0[0] = 1u;                                        // count=1, user desc
    g0[1] = lds_base;                                  // lds_addr (bytes)
    g0[2] = (unsigned int)ga;                          // global_addr[31:0]
    g0[3] = (unsigned int)(ga >> 32) | (2u << 30);     // ga[56:32] | type=2

    // D# group 1 (256b): data_size=4B, 1-D tile: tile_dim0 = 2048
    i32x8 g1;
    g1[0] = 0x00020000;                                // data_size=2 (4B)
    g1[1] = (int)((L_ & 0xFFFF) << 16);                // tensor_dim0 lo16
    g1[2] = (int)((1u << 16) | ((L_ >> 16) & 0xFFFF)); // tensor_dim1=1 | td0 hi
    g1[3] = (int)((L_ & 0xFFFF) << 16);                // tile_dim0 = 2048
    g1[4] = 0;                                         // tile_dim1/2 unused
    g1[5] = L_;                                        // tensor_dim0_stride
    g1[6] = 0;
    g1[7] = 0;

    i32x4 z4 = {0, 0, 0, 0};                           // groups 2/3 unused (1-D)
    i32x8 z8 = {0, 0, 0, 0, 0, 0, 0, 0};               // extra group (unused)
    __builtin_amdgcn_tensor_load_to_lds(g0, g1, z4, z4, z8, 0);

#if __has_builtin(__builtin_amdgcn_s_wait_tensorcnt)
    __builtin_amdgcn_s_wait_tensorcnt(0);
#else
    asm volatile("s_wait_tensorcnt 0" ::: "memory");
#endif
    asm volatile("" ::: "memory");                     // fence LDS reads below
  }
#endif

  float vmax = -3.402823466e+38f;
  float vsum = 0.0f;

#if HAVE_TDM
  const float4* p = (const float4*)&lds[wave * L_];
#else
  const float4* p = (const float4*)(x + (size_t)row * L_);
#endif
#pragma unroll
  for (int j = 0; j < L_ / (32 * 4); ++j) {            // 16 x float4 per lane
    float4 v = p[j * 32 + lane];
    vmax = fmaxf(fmaxf(vmax, v.x), fmaxf(v.y, fmaxf(v.z, v.w)));
    vsum += (v.x + v.y) + (v.z + v.w);
  }

  // wave32 tree reduction
#pragma unroll
  for (int off = 16; off >= 1; off >>= 1) {
    vmax = fmaxf(vmax, __shfl_xor(vmax, off, 32));
    vsum += __shfl_xor(vsum, off, 32);
  }
  if (lane == 0) {
    mx[row] = vmax;
    av[row] = vsum * (1.0f / (float)L_);
  }
}

// ---------------------------------------------------------------------------
// Kernel 2: per-batch fused 4-head channel-stencil + sigmoid + softmax combine
//   s = mx + av ;  att_h = conv1d(s, w_h, SAME/zero-pad over C)
//   out = mx * sum_h softmax(wv)[h] * sigmoid(att_h)
// ---------------------------------------------------------------------------
__global__ __launch_bounds__(256) void combine_kernel(
    const float* __restrict__ mx, const float* __restrict__ av,
    const float* __restrict__ w0, const float* __restrict__ w1,
    const float* __restrict__ w2, const float* __restrict__ w3,
    const float* __restrict__ wv, float* __restrict__ out) {
  __shared__ float s_sum[C_];
  __shared__ float s_mx[C_];
  const int b = (int)blockIdx.x;

  for (int c = (int)threadIdx.x; c < C_; c += 256) {
    float m = mx[b * C_ + c];
    s_mx[c] = m;
    s_sum[c] = m + av[b * C_ + c];
  }
  __syncthreads();

  // softmax over the 4 head weights (tiny; computed per-thread)
  float v0 = wv[0], v1 = wv[1], v2 = wv[2], v3 = wv[3];
  float wm = fmaxf(fmaxf(v0, v1), fmaxf(v2, v3));
  float e0 = expf(v0 - wm), e1 = expf(v1 - wm);
  float e2 = expf(v2 - wm), e3 = expf(v3 - wm);
  float inv = 1.0f / (e0 + e1 + e2 + e3);
  float sw0 = e0 * inv, sw1 = e1 * inv, sw2 = e2 * inv, sw3 = e3 * inv;

  float k0[3], k1[5], k2[7], k3[9];
#pragma unroll
  for (int j = 0; j < 3; ++j) k0[j] = w0[j];
#pragma unroll
  for (int j = 0; j < 5; ++j) k1[j] = w1[j];
#pragma unroll
  for (int j = 0; j < 7; ++j) k2[j] = w2[j];
#pragma unroll
  for (int j = 0; j < 9; ++j) k3[j] = w3[j];

  for (int c = (int)threadIdx.x; c < C_; c += 256) {
    float t[9];  // taps s[c-4 .. c+4], zero-padded
#pragma unroll
    for (int d = 0; d < 9; ++d) {
      int idx = c - 4 + d;
      t[d] = (idx >= 0 && idx < C_) ? s_sum[idx] : 0.0f;
    }
    float a0 = 0.f, a1 = 0.f, a2 = 0.f, a3 = 0.f;
#pragma unroll
    for (int j = 0; j < 3; ++j) a0 += k0[j] * t[j + 3];  // pad 1 -> center 4
#pragma unroll
    for (int j = 0; j < 5; ++j) a1 += k1[j] * t[j + 2];  // pad 2
#pragma unroll
    for (int j = 0; j < 7; ++j) a2 += k2[j] * t[j + 1];  // pad 3
#pragma unroll
    for (int j = 0; j < 9; ++j) a3 += k3[j] * t[j];      // pad 4

    float g = sw0 / (1.0f + expf(-a0)) + sw1 / (1.0f + expf(-a1)) +
              sw2 / (1.0f + expf(-a2)) + sw3 / (1.0f + expf(-a3));
    out[b * C_ + c] = s_mx[c] * g;
  }
}

// ---------------------------------------------------------------------------
extern "C" void kernel_launch(void* const* d_in, const int* in_sizes, int n_in,
                              void* d_out, int out_size, void* d_ws, size_t ws_size,
                              hipStream_t stream) {
  (void)in_sizes; (void)n_in; (void)out_size; (void)ws_size;
  const float* x  = (const float*)d_in[0];
  const float* w0 = (const float*)d_in[1];
  const float* w1 = (const float*)d_in[2];
  const float* w2 = (const float*)d_in[3];
  const float* w3 = (const float*)d_in[4];
  const float* wv = (const float*)d_in[5];

  float* mx = (float*)d_ws;        // ROWS floats
  float* av = mx + ROWS;           // ROWS floats (total 256 KB scratch)

  pool_kernel<<<ROWS / WAVES_PER_BLOCK, THREADS, 0, stream>>>(x, mx, av);
  combine_kernel<<<B_, 256, 0, stream>>>(mx, av, w0, w1, w2, w3, wv,
                                         (float*)d_out);
}